// RegionProposalNetwork_11742440587265
// MI455X (gfx1250) — compile-verified
//
#include <hip/hip_runtime.h>
#include <math.h>

typedef float v2f __attribute__((ext_vector_type(2)));
typedef float v8f __attribute__((ext_vector_type(8)));

// ---------------- problem constants ----------------
constexpr int BB = 4, CC = 256, HIDN = 256, AA = 15, HH = 100, WW = 152;
constexpr int NPIX  = HH * WW;          // 15200
constexpr int PH = HH + 2, PW = WW + 2; // 102, 154
constexpr int PHW   = PH * PW;          // 15708
constexpr int K3    = CC * 9;           // 2304
constexpr int NTILE = NPIX / 16;        // 950 (exact)
constexpr int MT1   = HIDN / 16;        // 16
constexpr int M2    = 80;               // 75 head rows padded to 80
constexpr int MT2   = M2 / 16;          // 5
constexpr int NANCH = NPIX * AA;        // 228000
constexpr int KTOP  = 2000;
constexpr int NWRD  = 63;               // ceil(2000/32)
constexpr int NS    = 262144;           // 2^18 >= NANCH
constexpr float NEGV = -1e30f;

// ---------------- workspace layout (bytes) ----------------
// Region A: Xpad -> (dead) -> head -> (dead) -> sort/NMS buffers
constexpr size_t REGA_SZ  = (size_t)BB * CC * PHW * 4;        // 64,339,968
constexpr size_t XPAD_OFF = 0;
constexpr size_t HEAD_OFF = 0;                                 // 19,456,000 used
constexpr size_t KEYS_OFF = 0;
constexpr size_t VALS_OFF = KEYS_OFF + (size_t)BB * NS * 4;
constexpr size_t SUPP_OFF = VALS_OFF + (size_t)BB * NS * 4;
constexpr size_t TOPB_OFF = SUPP_OFF + (size_t)BB * KTOP * NWRD * 4;
constexpr size_t TOPS_OFF = TOPB_OFF + (size_t)BB * KTOP * 16;
constexpr size_t AREA_OFF = TOPS_OFF + (size_t)BB * KTOP * 4;
// Region B: t -> (dead) -> boxes + scores
constexpr size_t B_OFF    = REGA_SZ;
constexpr size_t T_OFF    = B_OFF;                             // 62,259,200
constexpr size_t BOX_OFF  = B_OFF;
constexpr size_t SCO_OFF  = BOX_OFF + (size_t)BB * NANCH * 16;
constexpr size_t REGB_SZ  = (size_t)BB * HIDN * NPIX * 4;
// Region C: transformed weights
constexpr size_t C_OFF    = B_OFF + REGB_SZ;
constexpr size_t WC3_OFF  = C_OFF;                             // [256][9][256] f32
constexpr size_t WCAT_OFF = WC3_OFF + (size_t)HIDN * K3 * 4;
constexpr size_t BCAT_OFF = WCAT_OFF + (size_t)M2 * HIDN * 4;

__device__ __forceinline__ v8f wmma_f32(v2f a, v2f b, v8f c) {
  return __builtin_amdgcn_wmma_f32_16x16x4_f32(false, a, false, b, (short)0, c,
                                               false, false);
}

// ---------------- prep: weight transforms ----------------
__global__ void k_prep_w(const float* __restrict__ conv_w,
                         const float* __restrict__ obj_w,
                         const float* __restrict__ obj_b,
                         const float* __restrict__ bbox_w,
                         const float* __restrict__ bbox_b,
                         float* __restrict__ Wc3,   // [HID][9][256]
                         float* __restrict__ Wcat,  // [80][256]
                         float* __restrict__ bcat)  // [80]
{
  int i = blockIdx.x * 256 + threadIdx.x;
  const int NW3 = HIDN * K3;          // 589824
  const int NWC = M2 * HIDN;          // 20480
  if (i < NW3) {
    int m = i / K3, rem = i - m * K3;
    int j = rem >> 8, c = rem & 255;  // rem = j*256 + c
    Wc3[i] = conv_w[(size_t)m * K3 + c * 9 + j];
  } else if (i < NW3 + NWC) {
    int k = i - NW3;
    int r = k >> 8, c = k & 255;
    float v = 0.f;
    if (r < 15)      v = obj_w[r * HIDN + c];
    else if (r < 75) v = bbox_w[(r - 15) * HIDN + c];
    Wcat[k] = v;
  } else if (i < NW3 + NWC + M2) {
    int r = i - NW3 - NWC;
    float v = 0.f;
    if (r < 15)      v = obj_b[r];
    else if (r < 75) v = bbox_b[r - 15];
    bcat[r] = v;
  }
}

// ---------------- zero-padded feature copy ----------------
__global__ void k_pad(const float* __restrict__ feat, float* __restrict__ Xpad) {
  size_t i = (size_t)blockIdx.x * 256 + threadIdx.x;  // BB*CC*PHW total (exact)
  int px = (int)(i % PW);
  size_t r = i / PW;
  int py = (int)(r % PH);
  size_t bc = r / PH;
  float v = 0.f;
  if (py >= 1 && py <= HH && px >= 1 && px <= WW)
    v = feat[bc * NPIX + (size_t)(py - 1) * WW + (px - 1)];
  Xpad[i] = v;
}

// ---------------- 3x3 conv + bias + relu via WMMA ----------------
__global__ __launch_bounds__(128)
void k_conv3(const float* __restrict__ Xpad, const float* __restrict__ Wc3,
             const float* __restrict__ bias, float* __restrict__ t_out) {
  int wid  = blockIdx.x * 4 + (threadIdx.x >> 5);  // 60800 waves
  int lane = threadIdx.x & 31;
  int lo = lane & 15, hi = lane >> 4;
  int nt = wid % NTILE;
  int rest = wid / NTILE;
  int mt = rest & 15;
  int b  = rest >> 4;
  int pix = nt * 16 + lo;
  int y = pix / WW, x = pix - y * WW;

  const float* arow = Wc3 + (size_t)(mt * 16 + lo) * K3 + 2 * hi;
  const float* xb   = Xpad + (size_t)b * CC * PHW + (size_t)(2 * hi) * PHW;

  v8f acc = {0.f, 0.f, 0.f, 0.f, 0.f, 0.f, 0.f, 0.f};
  for (int j = 0; j < 9; ++j) {
    int ky = j / 3, kx = j - 3 * (j / 3);
    const float* bp = xb + (size_t)(y + ky) * PW + (x + kx);
    const float* ap = arow + j * 256;
#pragma unroll 8
    for (int c4 = 0; c4 < 256; c4 += 4) {
      v2f a = *(const v2f*)(ap + c4);
      v2f bv;
      bv.x = bp[0];
      bv.y = bp[PHW];
      bp += 4 * PHW;
      acc = wmma_f32(a, bv, acc);
    }
  }
  int ch0 = mt * 16 + 8 * hi;
  float* outp = t_out + (size_t)b * HIDN * NPIX + pix;
#pragma unroll
  for (int g = 0; g < 8; ++g) {
    int ch = ch0 + g;
    float v = acc[g] + bias[ch];
    outp[(size_t)ch * NPIX] = v > 0.f ? v : 0.f;
  }
}

// ---------------- fused 1x1 convs (obj + bbox) via WMMA ----------------
__global__ __launch_bounds__(128)
void k_head(const float* __restrict__ t_in, const float* __restrict__ Wcat,
            const float* __restrict__ bcat, float* __restrict__ head) {
  int wid  = blockIdx.x * 4 + (threadIdx.x >> 5);  // 19000 waves
  int lane = threadIdx.x & 31;
  int lo = lane & 15, hi = lane >> 4;
  int nt = wid % NTILE;
  int rest = wid / NTILE;
  int mt = rest % MT2;
  int b  = rest / MT2;
  int pix = nt * 16 + lo;

  const float* arow = Wcat + (size_t)(mt * 16 + lo) * HIDN + 2 * hi;
  const float* bp   = t_in + ((size_t)b * HIDN + 2 * hi) * NPIX + pix;

  v8f acc = {0.f, 0.f, 0.f, 0.f, 0.f, 0.f, 0.f, 0.f};
#pragma unroll 8
  for (int c4 = 0; c4 < 256; c4 += 4) {
    v2f a = *(const v2f*)(arow + c4);
    v2f bv;
    bv.x = bp[0];
    bv.y = bp[NPIX];
    bp += 4 * NPIX;
    acc = wmma_f32(a, bv, acc);
  }
  int ch0 = mt * 16 + 8 * hi;
  float* outp = head + (size_t)b * M2 * NPIX + pix;
#pragma unroll
  for (int g = 0; g < 8; ++g) {
    int ch = ch0 + g;
    outp[(size_t)ch * NPIX] = acc[g] + bcat[ch];
  }
}

// ---------------- decode + clip + min-size mask ----------------
__global__ void k_decode(const float* __restrict__ head,
                         const float* __restrict__ anchors,
                         float* __restrict__ boxes, float* __restrict__ scores) {
  int i = blockIdx.x * 256 + threadIdx.x;
  if (i >= BB * NANCH) return;
  int b = i / NANCH, n = i - b * NANCH;
  int p = n / AA, a = n - AA * p;
  const float* hp = head + (size_t)b * M2 * NPIX + p;
  float s  = hp[(size_t)a * NPIX];
  float d0 = hp[(size_t)(15 + a * 4 + 0) * NPIX];
  float d1 = hp[(size_t)(15 + a * 4 + 1) * NPIX];
  float d2 = hp[(size_t)(15 + a * 4 + 2) * NPIX];
  float d3 = hp[(size_t)(15 + a * 4 + 3) * NPIX];
  float a0 = anchors[4 * n + 0], a1 = anchors[4 * n + 1];
  float a2 = anchors[4 * n + 2], a3 = anchors[4 * n + 3];
  float wa = a2 - a0, ha = a3 - a1;
  float cxa = a0 + 0.5f * wa, cya = a1 + 0.5f * ha;
  float cx = cxa + d0 * wa, cy = cya + d1 * ha;
  float w = wa * expf(d2), h = ha * expf(d3);
  float x1 = cx - 0.5f * w, y1 = cy - 0.5f * h;
  float x2 = cx + 0.5f * w, y2 = cy + 0.5f * h;
  x1 = fminf(fmaxf(x1, 0.f), 1216.f);
  y1 = fminf(fmaxf(y1, 0.f), 800.f);
  x2 = fminf(fmaxf(x2, 0.f), 1216.f);
  y2 = fminf(fmaxf(y2, 0.f), 800.f);
  bool valid = ((x2 - x1) >= 0.001f) && ((y2 - y1) >= 0.001f);
  size_t o = (size_t)(b * NANCH + n) * 4;
  boxes[o + 0] = x1; boxes[o + 1] = y1; boxes[o + 2] = x2; boxes[o + 3] = y2;
  scores[(size_t)b * NANCH + n] = valid ? s : NEGV;
}

// ---------------- bitonic sort (descending by key) ----------------
__global__ void k_sort_init(const float* __restrict__ scores,
                            float* __restrict__ keys, int* __restrict__ vals) {
  int i = blockIdx.x * 256 + threadIdx.x;  // BB*NS exact
  int b = i >> 18, j = i & (NS - 1);
  keys[i] = (j < NANCH) ? scores[(size_t)b * NANCH + j] : -INFINITY;
  vals[i] = j;
}

__global__ void k_bstep(float* __restrict__ keys, int* __restrict__ vals,
                        int k, int j) {
  int i = blockIdx.x * 256 + threadIdx.x;  // 0..NS-1
  int b = blockIdx.y;
  float* kb = keys + (size_t)b * NS;
  int*   vb = vals + (size_t)b * NS;
  int ixj = i ^ j;
  if (ixj > i) {
    float a = kb[i], c = kb[ixj];
    bool up = ((i & k) == 0);            // overall descending
    if (up ? (a < c) : (a > c)) {
      kb[i] = c; kb[ixj] = a;
      int t = vb[i]; vb[i] = vb[ixj]; vb[ixj] = t;
    }
  }
}

__global__ __launch_bounds__(1024)
void k_blocal(float* __restrict__ keys, int* __restrict__ vals,
              int k, int jstart) {
  __shared__ float sk[2048];
  __shared__ int   sv[2048];
  int b = blockIdx.y;
  int base = blockIdx.x * 2048;
  float* kb = keys + (size_t)b * NS;
  int*   vb = vals + (size_t)b * NS;
  int t = threadIdx.x;
  sk[t] = kb[base + t];          sv[t] = vb[base + t];
  sk[t + 1024] = kb[base + t + 1024];
  sv[t + 1024] = vb[base + t + 1024];
  __syncthreads();
  for (int j = jstart; j > 0; j >>= 1) {
    int li = ((t & ~(j - 1)) << 1) | (t & (j - 1));
    int gi = base + li;
    float a = sk[li], c = sk[li + j];
    bool up = ((gi & k) == 0);
    if (up ? (a < c) : (a > c)) {
      sk[li] = c; sk[li + j] = a;
      int tv = sv[li]; sv[li] = sv[li + j]; sv[li + j] = tv;
    }
    __syncthreads();
  }
  kb[base + t] = sk[t];          vb[base + t] = sv[t];
  kb[base + t + 1024] = sk[t + 1024];
  vb[base + t + 1024] = sv[t + 1024];
}

// ---------------- gather top-K ----------------
__global__ void k_gather(const float* __restrict__ keys,
                         const int* __restrict__ vals,
                         const float* __restrict__ boxes,
                         float* __restrict__ topb, float* __restrict__ tops,
                         float* __restrict__ area) {
  int i = blockIdx.x * 256 + threadIdx.x;
  if (i >= BB * KTOP) return;
  int b = i / KTOP, t = i - b * KTOP;
  int idx = vals[(size_t)b * NS + t];
  const float* bx = boxes + ((size_t)b * NANCH + idx) * 4;
  float x1 = bx[0], y1 = bx[1], x2 = bx[2], y2 = bx[3];
  topb[(size_t)i * 4 + 0] = x1;
  topb[(size_t)i * 4 + 1] = y1;
  topb[(size_t)i * 4 + 2] = x2;
  topb[(size_t)i * 4 + 3] = y2;
  tops[i] = keys[(size_t)b * NS + t];
  area[i] = (x2 - x1) * (y2 - y1);
}

// ---------------- IoU suppression bitmask ----------------
__global__ void k_iou(const float* __restrict__ topb,
                      const float* __restrict__ area,
                      unsigned int* __restrict__ supp) {
  int gid = blockIdx.x * 256 + threadIdx.x;
  if (gid >= BB * KTOP * NWRD) return;
  int w = gid % NWRD;
  int r = gid / NWRD;
  int i = r % KTOP, b = r / KTOP;
  const float* tb = topb + (size_t)b * KTOP * 4;
  const float* ar = area + (size_t)b * KTOP;
  float ix1 = tb[i * 4 + 0], iy1 = tb[i * 4 + 1];
  float ix2 = tb[i * 4 + 2], iy2 = tb[i * 4 + 3];
  float ai = ar[i];
  unsigned int m = 0;
  int j0 = w * 32;
  for (int jj = 0; jj < 32; ++jj) {
    int j = j0 + jj;
    if (j < KTOP && j > i) {
      float xx1 = fmaxf(ix1, tb[j * 4 + 0]);
      float yy1 = fmaxf(iy1, tb[j * 4 + 1]);
      float xx2 = fminf(ix2, tb[j * 4 + 2]);
      float yy2 = fminf(iy2, tb[j * 4 + 3]);
      float iw = fmaxf(xx2 - xx1, 0.f);
      float ih = fmaxf(yy2 - yy1, 0.f);
      float inter = iw * ih;
      float iou = inter / (ai + ar[j] - inter + 1e-9f);
      if (iou > 0.7f) m |= (1u << jj);
    }
  }
  supp[(size_t)(b * KTOP + i) * NWRD + w] = m;
}

// ---------------- sequential greedy reduce + compaction ----------------
__global__ __launch_bounds__(256)
void k_nms(const unsigned int* __restrict__ supp, const float* __restrict__ tops,
           const float* __restrict__ topb, float* __restrict__ out) {
  int b = blockIdx.x;
  int t = threadIdx.x;
  __shared__ int posArr[KTOP];
  __shared__ int cntSh;
  const float* ts = tops + (size_t)b * KTOP;

  if (t < 32) {
    // accumulator of suppressed bits: lane t owns words t and t+32
    unsigned int acc0 = 0, acc1 = 0;
    for (int bit = 0; bit < 32; ++bit) {
      int j = t * 32 + bit;               // < 1024 < KTOP
      if (!(ts[j] > -5.0e29f)) acc0 |= (1u << bit);
    }
    if (t < 31) {
      for (int bit = 0; bit < 32; ++bit) {
        int j = (t + 32) * 32 + bit;
        if (j < KTOP && !(ts[j] > -5.0e29f)) acc1 |= (1u << bit);
      }
    }
    int cnt = 0;
    const unsigned int* srow = supp + (size_t)b * KTOP * NWRD;
    for (int i = 0; i < KTOP; ++i) {
      int wi = i >> 5;
      unsigned int w0 = (unsigned int)__shfl((int)acc0, wi & 31);
      unsigned int w1 = (unsigned int)__shfl((int)acc1, (wi - 32) & 31);
      unsigned int word = (wi < 32) ? w0 : w1;
      bool keep = ((word >> (i & 31)) & 1u) == 0u;
      if (keep) {
        const unsigned int* row = srow + (size_t)i * NWRD;
        acc0 |= row[t];
        if (t < 31) acc1 |= row[t + 32];
      }
      if (t == 0) posArr[i] = keep ? cnt : -1;
      cnt += keep ? 1 : 0;
    }
    if (t == 0) cntSh = cnt;
  }
  __syncthreads();
  int cnt = cntSh;
  float4* ob4 = (float4*)(out + (size_t)b * KTOP * 4);
  const float4* tb4 = (const float4*)(topb + (size_t)b * KTOP * 4);
  for (int i = t; i < KTOP; i += 256) {
    int p = posArr[i];
    if (p >= 0) ob4[p] = tb4[i];
  }
  for (int s = cnt + t; s < KTOP; s += 256) {
    ob4[s] = make_float4(0.f, 0.f, 0.f, 0.f);
  }
}

// ---------------- host launcher ----------------
extern "C" void kernel_launch(void* const* d_in, const int* in_sizes, int n_in,
                              void* d_out, int out_size, void* d_ws,
                              size_t ws_size, hipStream_t stream) {
  const float* feature = (const float*)d_in[0];
  const float* conv_w  = (const float*)d_in[1];
  const float* conv_b  = (const float*)d_in[2];
  const float* obj_w   = (const float*)d_in[3];
  const float* obj_b   = (const float*)d_in[4];
  const float* bbox_w  = (const float*)d_in[5];
  const float* bbox_b  = (const float*)d_in[6];
  const float* anchors = (const float*)d_in[7];
  float* out = (float*)d_out;

  char* ws = (char*)d_ws;
  float* Xpad  = (float*)(ws + XPAD_OFF);
  float* head  = (float*)(ws + HEAD_OFF);
  float* keys  = (float*)(ws + KEYS_OFF);
  int*   vals  = (int*)  (ws + VALS_OFF);
  unsigned int* supp = (unsigned int*)(ws + SUPP_OFF);
  float* topb  = (float*)(ws + TOPB_OFF);
  float* tops  = (float*)(ws + TOPS_OFF);
  float* area  = (float*)(ws + AREA_OFF);
  float* tbuf  = (float*)(ws + T_OFF);
  float* boxes = (float*)(ws + BOX_OFF);
  float* scores= (float*)(ws + SCO_OFF);
  float* Wc3   = (float*)(ws + WC3_OFF);
  float* Wcat  = (float*)(ws + WCAT_OFF);
  float* bcat  = (float*)(ws + BCAT_OFF);

  // 0) weight transforms + padded feature
  {
    int total = HIDN * K3 + M2 * HIDN + M2;
    k_prep_w<<<(total + 255) / 256, 256, 0, stream>>>(conv_w, obj_w, obj_b,
                                                      bbox_w, bbox_b, Wc3, Wcat,
                                                      bcat);
  }
  {
    size_t total = (size_t)BB * CC * PHW;   // exact multiple of 256
    k_pad<<<(unsigned)(total / 256), 256, 0, stream>>>(feature, Xpad);
  }
  // 1) 3x3 conv + relu  (WMMA f32)
  k_conv3<<<BB * MT1 * NTILE / 4, 128, 0, stream>>>(Xpad, Wc3, conv_b, tbuf);
  // 2) fused 1x1 heads  (WMMA f32)
  k_head<<<BB * MT2 * NTILE / 4, 128, 0, stream>>>(tbuf, Wcat, bcat, head);
  // 3) decode boxes
  k_decode<<<(BB * NANCH + 255) / 256, 256, 0, stream>>>(head, anchors, boxes,
                                                         scores);
  // 4) full bitonic sort (descending) of (score, idx) per image
  k_sort_init<<<BB * NS / 256, 256, 0, stream>>>(scores, keys, vals);
  for (int k = 2; k <= NS; k <<= 1) {
    int j = k >> 1;
    for (; j > 1024; j >>= 1) {
      k_bstep<<<dim3(NS / 256, BB), 256, 0, stream>>>(keys, vals, k, j);
    }
    k_blocal<<<dim3(NS / 2048, BB), 1024, 0, stream>>>(keys, vals, k, j);
  }
  // 5) gather top-K, areas
  k_gather<<<(BB * KTOP + 255) / 256, 256, 0, stream>>>(keys, vals, boxes, topb,
                                                        tops, area);
  // 6) suppression bitmask matrix
  k_iou<<<(BB * KTOP * NWRD + 255) / 256, 256, 0, stream>>>(topb, area, supp);
  // 7) sequential greedy NMS reduce + compact/zero output
  k_nms<<<BB, 256, 0, stream>>>(supp, tops, topb, out);
}